// HPLoss_72808285602050
// MI455X (gfx1250) — compile-verified
//
#include <hip/hip_runtime.h>
#include <hip/hip_bf16.h>
#include <math.h>

// Sizes from the reference
#define BS 1024
#define NP 256          // NB_PROXIES
#define D 128           // SZ_EMBED
#define T_BATCH (BS * 50)
#define T_LCA (NP * 50)

typedef __attribute__((ext_vector_type(2))) float v2f;
typedef __attribute__((ext_vector_type(8))) float v8f;

__device__ __forceinline__ float blk_reduce128(float v, float* red, int tid) {
    red[tid] = v;
    __syncthreads();
    for (int s = 64; s > 0; s >>= 1) {
        if (tid < s) red[tid] += red[tid + s];
        __syncthreads();
    }
    float r = red[0];
    __syncthreads();
    return r;
}

// lh = project(expmap0(clip(lcas))), plus per-row sum-of-squares.
__global__ void k_poincare(const float* __restrict__ lcas,
                           float* __restrict__ lh, float* __restrict__ sq_lh) {
    __shared__ float red[128];
    const int row = blockIdx.x;
    const int tid = threadIdx.x;
    float x = lcas[(size_t)row * D + tid];
    float ss = blk_reduce128(x * x, red, tid);
    float norm0 = sqrtf(ss);
    float xn = norm0 + 1e-5f;
    float s1 = fminf(1.0f, 2.3f / xn);          // CLIP_R
    float xc = x * s1;
    float xn2 = fmaxf(norm0 * s1, 1e-5f);
    const float sqrt_c = 0.31622776601683794f;  // sqrt(0.1)
    float th = tanhf(sqrt_c * xn2);
    float e = xc * (th / (sqrt_c * xn2));       // expmap0
    float en = fmaxf(th / sqrt_c, 1e-5f);       // ||e|| (uniform scale of xc)
    float maxnorm = (1.0f - 1e-3f) / sqrt_c;
    float ps = (en > maxnorm) ? (maxnorm / en) : 1.0f;   // project()
    float v = e * ps;
    lh[(size_t)row * D + tid] = v;
    float ss2 = blk_reduce128(v * v, red, tid);
    if (tid == 0) sq_lh[row] = ss2;
}

// Row sum-of-squares for z_s and t_s.
__global__ void k_rowsq(const float* __restrict__ z, const float* __restrict__ t,
                        float* __restrict__ sq_z, float* __restrict__ sq_t) {
    __shared__ float red[128];
    const int row = blockIdx.x;   // 0..2047
    const int tid = threadIdx.x;
    const bool isZ = row < BS;
    const int r = isZ ? row : row - BS;
    const float* src = isZ ? z : t;
    float x = src[(size_t)r * D + tid];
    float ss = blk_reduce128(x * x, red, tid);
    if (tid == 0) (isZ ? sq_z : sq_t)[r] = ss;
}

// One wave per 16x16 tile. Three GEMMs (z*lh^T, lh*lh^T, t*lh^T), K=128,
// done with V_WMMA_F32_16X16X4_F32, distance transform fused in the epilogue.
__global__ void k_dist_wmma(const float* __restrict__ z, const float* __restrict__ t,
                            const float* __restrict__ lh,
                            const float* __restrict__ sq_z, const float* __restrict__ sq_t,
                            const float* __restrict__ sq_lh,
                            float* __restrict__ cp1, float* __restrict__ cp2,
                            float* __restrict__ cp3) {
    const int tile = blockIdx.x;
    const float* X; const float* x2; float* out; int local;
    if (tile < 1024)      { X = z;  x2 = sq_z;  out = cp1; local = tile; }        // 64x16 tiles
    else if (tile < 1280) { X = lh; x2 = sq_lh; out = cp2; local = tile - 1024; } // 16x16 tiles
    else                  { X = t;  x2 = sq_t;  out = cp3; local = tile - 1280; }
    const int tileM = (local >> 4) << 4;
    const int tileN = (local & 15) << 4;

    const int lane = threadIdx.x;          // block == one wave32
    const int half = lane >> 4;            // 0: K pair {0,1}, 1: K pair {2,3}
    const int r = lane & 15;
    const float* aptr = X  + (size_t)(tileM + r) * D + 2 * half;
    const float* bptr = lh + (size_t)(tileN + r) * D + 2 * half;

    v8f c = {0.f, 0.f, 0.f, 0.f, 0.f, 0.f, 0.f, 0.f};
    for (int kk = 0; kk < D; kk += 4) {
        v2f a = *(const v2f*)(aptr + kk);
        v2f b = *(const v2f*)(bptr + kk);
        c = __builtin_amdgcn_wmma_f32_16x16x4_f32(false, a, false, b,
                                                  (short)0, c, false, false);
    }

    const float cc = 0.1f;                       // HYP_C
    const float sqrt_c = 0.31622776601683794f;
    const float two_over_sqrt_c = 6.324555320336759f;
    const float Y2 = sq_lh[tileN + r];
    for (int v = 0; v < 8; v++) {
        const int mo = tileM + v + half * 8;     // C/D VGPR layout
        const float dot = c[v];
        const float xy = -dot;                   // reference: xy = -(x . y)
        const float X2 = x2[mo];
        const float a_ = 1.0f + 2.0f * cc * xy + cc * Y2;
        const float b_ = 1.0f - cc * X2;
        const float denom = 1.0f + 2.0f * cc * xy + cc * cc * X2 * Y2;
        const float num_sq = a_ * a_ * X2 + 2.0f * a_ * b_ * xy + b_ * b_ * Y2;
        const float nrm = sqrtf(fmaxf(num_sq, 1e-12f)) / fabsf(denom + 1e-5f);
        float arg = sqrt_c * nrm;
        arg = fminf(fmaxf(arg, -1.0f + 1e-5f), 1.0f - 1e-5f);
        const float at = 0.5f * (log1pf(arg) - log1pf(-arg));   // artanh
        out[(size_t)mo * NP + tileN + r] = two_over_sqrt_c * at;
    }
}

__global__ void k_zero(float* out) {
    if (threadIdx.x == 0 && blockIdx.x == 0) out[0] = 0.0f;
}

// One wave per triplet. Forward gumbel_softmax_hard == one-hot(argmax(logits)),
// so sums collapse to single gathers at the two argmax proxies.
// Lane `l` owns p in [8l, 8l+8) so all row traffic is b128 loads; first-
// occurrence argmax is preserved (per-lane p increasing + smaller-index ties).
__global__ void k_trip(const float* __restrict__ cp, const int* __restrict__ trip,
                       const float* __restrict__ g, int T, float invT,
                       float* __restrict__ out) {
    const int wave = threadIdx.x >> 5;
    const int lane = threadIdx.x & 31;
    const int t = blockIdx.x * 8 + wave;
    if (t >= T) return;   // uniform per wave

    const int i = trip[t];
    const int j = trip[T + t];
    const int k = trip[2 * T + t];
    const float* di = cp + (size_t)i * NP;
    const float* dj = cp + (size_t)j * NP;
    const float* dk = cp + (size_t)k * NP;
    const float* g0 = g + (size_t)t * NP;
    const float* g1 = g + (size_t)T * NP + (size_t)t * NP;

    float b1 = -3.4e38f; int i1 = 0;
    float b2 = -3.4e38f; int i2 = 0;
#pragma unroll
    for (int u = 0; u < 2; u++) {
        const int q = lane * 2 + u;                      // float4 chunk index
        const float4 vi = ((const float4*)di)[q];
        const float4 vj = ((const float4*)dj)[q];
        const float4 vk = ((const float4*)dk)[q];
        const float4 ga = ((const float4*)g0)[q];
        const float4 gb = ((const float4*)g1)[q];
        float m[4]  = { fmaxf(vi.x, vj.x), fmaxf(vi.y, vj.y),
                        fmaxf(vi.z, vj.z), fmaxf(vi.w, vj.w) };
        float kk[4] = { vk.x, vk.y, vk.z, vk.w };
        float gav[4] = { ga.x, ga.y, ga.z, ga.w };
        float gbv[4] = { gb.x, gb.y, gb.z, gb.w };
#pragma unroll
        for (int e = 0; e < 4; e++) {
            const int p = q * 4 + e;
            const float l1 = -m[e] * 10.0f + gav[e];                 // 1/TAU = 10
            const float l2 = -fmaxf(kk[e], m[e]) * 10.0f + gbv[e];
            if (l1 > b1) { b1 = l1; i1 = p; }                        // keeps first max
            if (l2 > b2) { b2 = l2; i2 = p; }
        }
    }
    for (int off = 16; off >= 1; off >>= 1) {
        float ob = __shfl_xor(b1, off, 32); int oi = __shfl_xor(i1, off, 32);
        if (ob > b1 || (ob == b1 && oi < i1)) { b1 = ob; i1 = oi; }
        ob = __shfl_xor(b2, off, 32); oi = __shfl_xor(i2, off, 32);
        if (ob > b2 || (ob == b2 && oi < i2)) { b2 = ob; i2 = oi; }
    }
    if (lane == 0 && i1 != i2) {     // diff_lca gate
        const float hc = fmaxf(di[i1] - di[i2] + 0.1f, 0.0f)
                       + fmaxf(dj[i1] - dj[i2] + 0.1f, 0.0f)
                       + fmaxf(dk[i2] - dk[i1] + 0.1f, 0.0f);
        atomicAdd(out, hc * invT);
    }
}

extern "C" void kernel_launch(void* const* d_in, const int* in_sizes, int n_in,
                              void* d_out, int out_size, void* d_ws, size_t ws_size,
                              hipStream_t stream) {
    const float* z_s  = (const float*)d_in[0];
    const float* t_s  = (const float*)d_in[1];
    // d_in[2] = y (unused by the math)
    const float* lcas = (const float*)d_in[3];
    const int* trip1 = (const int*)d_in[4];
    const int* trip2 = (const int*)d_in[5];
    const int* trip3 = (const int*)d_in[6];
    const int* trip4 = (const int*)d_in[7];
    const float* g1 = (const float*)d_in[8];
    const float* g2 = (const float*)d_in[9];
    const float* g3 = (const float*)d_in[10];
    const float* g4 = (const float*)d_in[11];
    float* out = (float*)d_out;

    // Workspace layout (floats); total ~2.5 MB
    float* ws = (float*)d_ws;
    float* lh    = ws;                       // 256*128
    float* sq_lh = lh + NP * D;              // 256
    float* sq_z  = sq_lh + NP;               // 1024
    float* sq_t  = sq_z + BS;                // 1024
    float* cp1   = sq_t + BS;                // 1024*256  dist(z_s, lh)
    float* cp2   = cp1 + (size_t)BS * NP;    // 256*256   dist(lh, lh)
    float* cp3   = cp2 + (size_t)NP * NP;    // 1024*256  dist(t_s, lh)

    k_zero<<<1, 32, 0, stream>>>(out);
    k_poincare<<<NP, 128, 0, stream>>>(lcas, lh, sq_lh);
    k_rowsq<<<2 * BS, 128, 0, stream>>>(z_s, t_s, sq_z, sq_t);
    // 1024 + 256 + 1024 = 2304 16x16 tiles, one wave each
    k_dist_wmma<<<2304, 32, 0, stream>>>(z_s, t_s, lh, sq_z, sq_t, sq_lh,
                                         cp1, cp2, cp3);

    const float invTB = 1.0f / (float)T_BATCH;
    const float invTL = 1.0f / (float)T_LCA;
    k_trip<<<(T_BATCH + 7) / 8, 256, 0, stream>>>(cp1, trip1, g1, T_BATCH, invTB, out);
    k_trip<<<(T_LCA   + 7) / 8, 256, 0, stream>>>(cp2, trip2, g2, T_LCA,   invTL, out);
    k_trip<<<(T_BATCH + 7) / 8, 256, 0, stream>>>(cp3, trip3, g3, T_BATCH, invTB, out);
    k_trip<<<(T_LCA   + 7) / 8, 256, 0, stream>>>(cp2, trip4, g4, T_LCA,   invTL, out);
}